// XingLoss_31284541784235
// MI455X (gfx1250) — compile-verified
//
#include <hip/hip_runtime.h>
#include <math.h>

// ---- types ----
typedef __attribute__((ext_vector_type(2))) float v2f;
typedef __attribute__((ext_vector_type(8))) float v8f;
typedef __attribute__((ext_vector_type(4))) unsigned int v4u;
typedef __attribute__((ext_vector_type(4))) int v4i;
typedef __attribute__((address_space(1))) v4i gv4i;     // global int4
typedef __attribute__((address_space(3))) v4i lv4i;     // LDS int4
typedef __attribute__((address_space(1))) char gchar;
typedef __attribute__((address_space(3))) char lchar;

#define NPTS        1024
#define SEGN        256                 // N/4 segments per row
#define ROW_FLOATS  (NPTS * 2)          // 2048 floats = 8192 B per row
#define CHUNKS      385                 // ceil(6152 B / 16 B) -> stage 6160 B (within row)
#define THREADS     256

#if __has_builtin(__builtin_amdgcn_global_load_async_to_lds_b128)
#define HAVE_ASYNC_LDS 1
#else
#define HAVE_ASYNC_LDS 0
#endif

// Kernel 1: one block per row. Stage needed 6160 B of the row into LDS with
// async b128 loads, compute the 256 segment terms, WMMA-reduce to one partial
// per block.
__global__ __launch_bounds__(THREADS) void xing_rows(const float* __restrict__ x,
                                                     float* __restrict__ ws)
{
    __shared__ float tile[CHUNKS * 4];   // 6160 B staged row data
    __shared__ float term[THREADS];

    const int tid = threadIdx.x;
    const float* row = x + (size_t)blockIdx.x * ROW_FLOATS;

#if HAVE_ASYNC_LDS
    {
        gchar* g = (gchar*)row;                    // generic -> global AS cast
        lchar* l = (lchar*)tile;                   // generic -> LDS AS cast
        __builtin_amdgcn_global_load_async_to_lds_b128(
            (gv4i*)(g + (size_t)tid * 16), (lv4i*)(l + tid * 16), 0, 0);
        if (tid < CHUNKS - THREADS)
            __builtin_amdgcn_global_load_async_to_lds_b128(
                (gv4i*)(g + (size_t)(THREADS + tid) * 16),
                (lv4i*)(l + (THREADS + tid) * 16), 0, 0);
#if __has_builtin(__builtin_amdgcn_s_wait_asynccnt)
        __builtin_amdgcn_s_wait_asynccnt(0);
#else
        asm volatile("s_wait_asynccnt 0" ::: "memory");
#endif
    }
#else
    {
        const v4u* g = (const v4u*)row;
        v4u* l = (v4u*)tile;
        l[tid] = __builtin_nontemporal_load(g + tid);
        if (tid < CHUNKS - THREADS)
            l[THREADS + tid] = __builtin_nontemporal_load(g + THREADS + tid);
    }
#endif
    __syncthreads();

    // segment tid uses points 3t .. 3t+3 (float2 LDS reads, conflict-free:
    // 6t mod 64 is a bijection over even banks for t in a wave)
    {
        const v2f* pts = (const v2f*)tile;
        v2f p0 = pts[3 * tid + 0];
        v2f p1 = pts[3 * tid + 1];
        v2f p2 = pts[3 * tid + 2];
        v2f p3 = pts[3 * tid + 3];
        float v1x = p1.x - p0.x, v1y = p1.y - p0.y;
        float v2x = p2.x - p1.x, v2y = p2.y - p1.y;
        float v3x = p3.x - p2.x, v3y = p3.y - p2.y;
        float c12 = v1x * v2y - v1y * v2x;                 // sign(s12) == sign(c12)
        float c13 = v1x * v3y - v1y * v3x;
        float k   = sqrtf((v1x * v1x + v1y * v1y) * (v3x * v3x + v3y * v3y));
        float sina = c13 / k;
        term[tid] = (c12 >= 0.0f) ? fmaxf(-sina, 0.0f) : fmaxf(sina, 0.0f);
    }
    __syncthreads();

    // wave 0: reduce 256 terms. WMMA trick: A = ones(16x4) => D[m][n] = sum_k B[k][n],
    // so every one of the 256 values lands (replicated over rows) in the column sums.
    // Sum of D-VGPR0 over all 32 lanes counts every input exactly twice.
    if (tid < 32) {
        float s;
#if __has_builtin(__builtin_amdgcn_wmma_f32_16x16x4_f32)
        v8f acc = {};
        v2f ones; ones.x = 1.0f; ones.y = 1.0f;
#pragma unroll
        for (int i = 0; i < 4; ++i) {
            v2f b;
            b.x = term[i * 64 + tid];
            b.y = term[i * 64 + 32 + tid];
            acc = __builtin_amdgcn_wmma_f32_16x16x4_f32(
                false, ones, false, b, (short)0, acc, false, false);
        }
        s = acc[0];
#else
        s = 0.0f;
        for (int i = tid; i < THREADS; i += 32) s += term[i] * 2.0f; // keep 2x convention
#endif
        for (int off = 16; off > 0; off >>= 1) s += __shfl_xor(s, off, 32);
        if (tid == 0) ws[blockIdx.x] = s * 0.5f;
    }
}

// Kernel 2: deterministic final reduction of P partials -> scalar mean.
__global__ __launch_bounds__(256) void xing_final(const float* __restrict__ ws, int nblocks,
                                                  const int* __restrict__ scale,
                                                  float* __restrict__ out, float inv_pn)
{
    __shared__ float red[256];
    const int tid = threadIdx.x;
    float s = 0.0f;
    for (int i = tid; i < nblocks; i += 256) s += ws[i];
    red[tid] = s;
    __syncthreads();
    for (int w = 128; w > 0; w >>= 1) {
        if (tid < w) red[tid] += red[tid + w];
        __syncthreads();
    }
    if (tid == 0) out[0] = red[0] * (float)scale[0] * inv_pn;
}

extern "C" void kernel_launch(void* const* d_in, const int* in_sizes, int n_in,
                              void* d_out, int out_size, void* d_ws, size_t ws_size,
                              hipStream_t stream)
{
    const float* x   = (const float*)d_in[0];
    const int* scale = (const int*)d_in[1];
    const int P = in_sizes[0] / ROW_FLOATS;           // 16384

    float* ws = (float*)d_ws;                         // P floats of scratch
    xing_rows<<<P, THREADS, 0, stream>>>(x, ws);

    const float inv_pn = 1.0f / ((float)P * (float)SEGN);
    xing_final<<<1, 256, 0, stream>>>(ws, P, scale, (float*)d_out, inv_pn);
}